// CustomAttention_69166153335084
// MI455X (gfx1250) — compile-verified
//
#include <hip/hip_runtime.h>

// ---------------------------------------------------------------------------
// HeteroGAT (3 layers, 4 heads x 64) for MI455X / gfx1250.
// Dense GEMMs: V_WMMA_F32_16X16X4_F32 with the B (weight) panel staged into
// LDS via GLOBAL_LOAD_ASYNC_TO_LDS_B128 (ASYNCcnt) - removes redundant
// strided global B reads across the 8 waves of a block.
// Edge softmax/scatter uses L2-resident atomics (working set < 192MB L2).
// ---------------------------------------------------------------------------

typedef float v2f __attribute__((ext_vector_type(2)));
typedef float v8f __attribute__((ext_vector_type(8)));

#define HEADS       4
#define HID         64
#define GAT_N       256           // HEADS*HID
#define NEG_SLOPE   0.2f
#define MAXKEY_INIT 0x007FFFFFu   // fkey(-inf)
#define KC          32            // K-chunk staged in LDS (divides 64/128/256)

// order-preserving float<->uint key for atomicMax on floats
__device__ __forceinline__ unsigned fkey(float f) {
    unsigned u = __float_as_uint(f);
    return (u & 0x80000000u) ? ~u : (u | 0x80000000u);
}
__device__ __forceinline__ float funkey(unsigned k) {
    unsigned u = (k & 0x80000000u) ? (k & 0x7FFFFFFFu) : ~k;
    return __uint_as_float(u);
}

// ---------------------------------------------------------------------------
// OUT[M x N] = act(X[M x K] @ W[K x N] + bias)   (row-major, N in {64,256},
// K % 32 == 0). Block = 256 threads = 8 waves; each wave computes a 16x64
// output strip (4 f32 WMMA accumulators sharing one A fragment per k-step).
// The W k-panel (KC x N, contiguous in memory) is staged cooperatively into
// LDS with async b128 copies; B fragments then come from LDS (ds_load).
// Fragment layouts per CDNA5 ISA 7.12.2:
//   A 16x4 : lane<16 holds M=lane, K={0,1}; lane>=16 holds M=lane-16, K={2,3}
//   B 4x16 : lane<16 holds N=lane, K={0,1}; lane>=16 holds N=lane-16, K={2,3}
//   C/D    : vgpr v, lane<16 -> row v; lane>=16 -> row v+8; col = lane&15
// ---------------------------------------------------------------------------
__global__ __launch_bounds__(256) void gemm_wmma_f32(
    const float* __restrict__ X, const float* __restrict__ W,
    const float* __restrict__ bias, float* __restrict__ OUT,
    int M, int K, int N, int relu)
{
    __shared__ float sW[KC * 256];             // 32KB max (N=256)

    const int tid    = threadIdx.x;
    const int lane   = tid & 31;
    const int wv     = tid >> 5;               // wave index in block, 0..7
    const int strips = N >> 6;                 // 1 (N=64) or 4 (N=256)
    const int mtPerB = 8 / strips;             // mTiles covered per block
    const int mTile  = blockIdx.x * mtPerB + wv / strips;
    const int nStrip = wv % strips;
    const bool active = (mTile * 16) < M;

    const int mLane = mTile * 16 + (lane & 15);
    const int rowA  = (active && mLane < M) ? mLane : 0;
    const int kHalf = (lane >> 4) << 1;        // 0 or 2
    const int nCol  = nStrip * 64 + (lane & 15);

    // LDS byte offset of sW: low 32 bits of the shared-aperture address
    const unsigned ldsBase = (unsigned)(uintptr_t)&sW[0];

    v8f acc[4] = {};
    const float* Xr = X + (long)rowA * K;

    const int vec4PerThread = (KC * N) >> 10;  // (KC*N/4)/256 : 2 or 8

    for (int k0 = 0; k0 < K; k0 += KC) {
        // ---- cooperative async copy: W[k0 .. k0+KC) x N  ->  LDS ----------
        const float* src = W + (long)k0 * N;   // contiguous KC*N floats
        for (int i = 0; i < vec4PerThread; ++i) {
            const int v4 = i * 256 + tid;      // float4 index in chunk
            const unsigned ldsOff = ldsBase + (unsigned)(v4 << 4);
            const float* g = src + (v4 << 2);
            asm volatile("global_load_async_to_lds_b128 %0, %1, off"
                         :: "v"(ldsOff), "v"(g) : "memory");
        }
        asm volatile("s_wait_asynccnt 0" ::: "memory");
        __syncthreads();

        // ---- WMMA over the staged panel -----------------------------------
        if (active) {
            for (int kc = 0; kc < KC; kc += 4) {
                const int kk = kc + kHalf;
                v2f a = { Xr[k0 + kk], Xr[k0 + kk + 1] };
                const int sB = kk * N + nCol;
#pragma unroll
                for (int j = 0; j < 4; ++j) {
                    v2f b = { sW[sB + j * 16], sW[sB + N + j * 16] };
                    acc[j] = __builtin_amdgcn_wmma_f32_16x16x4_f32(
                        false, a, false, b, 0, acc[j], false, false);
                }
            }
        }
        __syncthreads();                       // before next chunk overwrites
    }
    if (!active) return;

    const int nIdx  = lane & 15;
    const int mBase = mTile * 16 + ((lane >> 4) << 3);
#pragma unroll
    for (int j = 0; j < 4; ++j) {
        const int col = nStrip * 64 + j * 16 + nIdx;
        const float bv = bias ? bias[col] : 0.0f;
#pragma unroll
        for (int v = 0; v < 8; ++v) {
            const int r = mBase + v;
            if (r < M) {
                float val = acc[j][v] + bv;
                if (relu) val = fmaxf(val, 0.0f);
                OUT[(long)r * N + col] = val;
            }
        }
    }
}

// es[n,h] = dot(H[n, h*64 : h*64+64], att[h])
__global__ __launch_bounds__(256) void node_scores(
    const float* __restrict__ H, const float* __restrict__ att,
    float* __restrict__ OUT, int n)
{
    int t = blockIdx.x * blockDim.x + threadIdx.x;
    if (t >= n * HEADS) return;
    const int node = t >> 2, h = t & 3;
    const float* hp = H + (long)node * GAT_N + h * HID;
    const float* ap = att + h * HID;
    float acc = 0.f;
#pragma unroll 8
    for (int c = 0; c < HID; ++c) acc += hp[c] * ap[c];
    OUT[t] = acc;
}

__global__ void fill_f32(float* p, int n, float v) {
    int t = blockIdx.x * blockDim.x + threadIdx.x;
    if (t < n) p[t] = v;
}
__global__ void fill_u32(unsigned* p, int n, unsigned v) {
    int t = blockIdx.x * blockDim.x + threadIdx.x;
    if (t < n) p[t] = v;
}

// pass 1: segment-max of leaky_relu(es[src]+ed[dst]) via ordered-uint atomicMax
__global__ __launch_bounds__(256) void edge_max_k(
    const int* __restrict__ src, const int* __restrict__ dst,
    const float* __restrict__ ES, const float* __restrict__ ED,
    unsigned* __restrict__ MXk, int E)
{
    int t = blockIdx.x * blockDim.x + threadIdx.x;
    if (t >= E * HEADS) return;
    const int e = t >> 2, h = t & 3;
    const int s = src[e], d = dst[e];
    float v = ES[s * 4 + h] + ED[d * 4 + h];
    v = (v > 0.f) ? v : NEG_SLOPE * v;
    atomicMax(&MXk[d * 4 + h], fkey(v));
}

// decode keys in place; nodes with no in-edges get m = 0
__global__ void decode_max(float* MX, int n) {
    int t = blockIdx.x * blockDim.x + threadIdx.x;
    if (t >= n) return;
    const unsigned k = __float_as_uint(MX[t]);
    MX[t] = (k == MAXKEY_INIT) ? 0.0f : funkey(k);
}

// pass 2: denom[dst,h] += exp(e - m[dst,h])
__global__ __launch_bounds__(256) void edge_denom_k(
    const int* __restrict__ src, const int* __restrict__ dst,
    const float* __restrict__ ES, const float* __restrict__ ED,
    const float* __restrict__ MX, float* __restrict__ DN, int E)
{
    int t = blockIdx.x * blockDim.x + threadIdx.x;
    if (t >= E * HEADS) return;
    const int e = t >> 2, h = t & 3;
    const int s = src[e], d = dst[e];
    float v = ES[s * 4 + h] + ED[d * 4 + h];
    v = (v > 0.f) ? v : NEG_SLOPE * v;
    atomicAdd(&DN[d * 4 + h], __expf(v - MX[d * 4 + h]));
}

// pass 3: OUT[dst, h*64+c] += alpha * HS[src, h*64+c]; one block per edge,
// 256 threads = (head, channel). alpha recomputed from L2-resident scalars.
__global__ __launch_bounds__(256) void edge_scatter_k(
    const int* __restrict__ src, const int* __restrict__ dst,
    const float* __restrict__ HS,
    const float* __restrict__ ES, const float* __restrict__ ED,
    const float* __restrict__ MX, const float* __restrict__ DN,
    float* __restrict__ OUT, int E)
{
    const int e = blockIdx.x;
    if (e >= E) return;
    const int idx = threadIdx.x;          // 0..255
    const int h = idx >> 6, c = idx & 63;
    const int s = src[e], d = dst[e];
    float v = ES[s * 4 + h] + ED[d * 4 + h];
    v = (v > 0.f) ? v : NEG_SLOPE * v;
    const float ex = __expf(v - MX[d * 4 + h]);
    const float alpha = ex / (DN[d * 4 + h] + 1e-16f);
    atomicAdd(&OUT[(long)d * GAT_N + h * HID + c],
              HS[(long)s * GAT_N + h * HID + c] * alpha);
}

// X[n*256+j] = relu(X[n*256+j] + bias[j]) in place
__global__ __launch_bounds__(256) void bias_relu_k(
    float* __restrict__ X, const float* __restrict__ bias, int total)
{
    int t = blockIdx.x * blockDim.x + threadIdx.x;
    if (t >= total) return;
    X[t] = fmaxf(X[t] + bias[t & (GAT_N - 1)], 0.0f);
}

// ---------------------------------------------------------------------------
static inline int cdiv(long a, int b) { return (int)((a + b - 1) / b); }

static inline void launch_gemm(const float* X, const float* W, const float* bias,
                               float* OUT, int M, int K, int N, int relu,
                               hipStream_t s)
{
    const int strips  = N / 64;
    const int mtPerB  = 8 / strips;
    const int tilesM  = (M + 15) / 16;
    const int blocks  = (tilesM + mtPerB - 1) / mtPerB;
    gemm_wmma_f32<<<blocks, 256, 0, s>>>(X, W, bias, OUT, M, K, N, relu);
}

static void run_gat(const float* Xs, int n_src, const float* Xd, int n_dst, int din,
                    const int* src, const int* dst, int E,
                    const float* W, const float* as_, const float* ad_,
                    const float* bias,
                    float* HS, float* HD, float* ES, float* ED,
                    float* MX, float* DN, float* OUT, hipStream_t s)
{
    launch_gemm(Xs, W, nullptr, HS, n_src, din, GAT_N, 0, s);
    launch_gemm(Xd, W, nullptr, HD, n_dst, din, GAT_N, 0, s);
    node_scores<<<cdiv((long)n_src * HEADS, 256), 256, 0, s>>>(HS, as_, ES, n_src);
    node_scores<<<cdiv((long)n_dst * HEADS, 256), 256, 0, s>>>(HD, ad_, ED, n_dst);
    fill_u32<<<cdiv((long)n_dst * HEADS, 256), 256, 0, s>>>((unsigned*)MX, n_dst * HEADS, MAXKEY_INIT);
    fill_f32<<<cdiv((long)n_dst * HEADS, 256), 256, 0, s>>>(DN, n_dst * HEADS, 0.0f);
    fill_f32<<<cdiv((long)n_dst * GAT_N, 256), 256, 0, s>>>(OUT, n_dst * GAT_N, 0.0f);
    edge_max_k<<<cdiv((long)E * HEADS, 256), 256, 0, s>>>(src, dst, ES, ED, (unsigned*)MX, E);
    decode_max<<<cdiv((long)n_dst * HEADS, 256), 256, 0, s>>>(MX, n_dst * HEADS);
    edge_denom_k<<<cdiv((long)E * HEADS, 256), 256, 0, s>>>(src, dst, ES, ED, MX, DN, E);
    edge_scatter_k<<<E, 256, 0, s>>>(src, dst, HS, ES, ED, MX, DN, OUT, E);
    bias_relu_k<<<cdiv((long)n_dst * GAT_N, 256), 256, 0, s>>>(OUT, bias, n_dst * GAT_N);
}

extern "C" void kernel_launch(void* const* d_in, const int* in_sizes, int n_in,
                              void* d_out, int out_size, void* d_ws, size_t ws_size,
                              hipStream_t stream)
{
    const int F_IN = 128;
    const float* x_user = (const float*)d_in[0];
    const float* x_item = (const float*)d_in[1];
    const int*   e_ui   = (const int*)d_in[2];   // [0..E): user src, [E..2E): item dst
    const int*   e_iu   = (const int*)d_in[3];   // [0..E): item src, [E..2E): user dst
    const int n_user = in_sizes[0] / F_IN;
    const int n_item = in_sizes[1] / F_IN;
    const int E_ui   = in_sizes[2] / 2;
    const int E_iu   = in_sizes[3] / 2;

    // params flattened in insertion order:
    //   pre_u[W,b], pre_i[W,b],
    //   layers[0..2]: ui[W, att_src, att_dst, b], iu[W, att_src, att_dst, b],
    //   post_u[W0,b0,W2,b3], post_i[W0,b0,W2,b3]
    const float* pre_uW = (const float*)d_in[4];
    const float* pre_ub = (const float*)d_in[5];
    const float* pre_iW = (const float*)d_in[6];
    const float* pre_ib = (const float*)d_in[7];
    const float* post_u[4] = { (const float*)d_in[32], (const float*)d_in[33],
                               (const float*)d_in[34], (const float*)d_in[35] };
    const float* post_i[4] = { (const float*)d_in[36], (const float*)d_in[37],
                               (const float*)d_in[38], (const float*)d_in[39] };

    // workspace layout (floats)
    const int nmax = (n_user > n_item) ? n_user : n_item;
    const size_t SZ = (size_t)nmax * GAT_N;
    float* ws = (float*)d_ws;
    float* U[2] = { ws,           ws + SZ };
    float* I[2] = { ws + 2 * SZ,  ws + 3 * SZ };
    float* HS = ws + 4 * SZ;
    float* HD = ws + 5 * SZ;
    float* ES = ws + 6 * SZ;
    float* ED = ES + (size_t)nmax * HEADS;
    float* MX = ED + (size_t)nmax * HEADS;
    float* DN = MX + (size_t)nmax * HEADS;

    // pre-projection: H = relu(x @ W + b), [N,128]@[128,64]
    launch_gemm(x_user, pre_uW, pre_ub, U[0], n_user, F_IN, HID, 1, stream);
    launch_gemm(x_item, pre_iW, pre_ib, I[0], n_item, F_IN, HID, 1, stream);

    for (int l = 0; l < 3; ++l) {
        const int din = (l == 0) ? HID : GAT_N;
        const int cur = l & 1, nxt = cur ^ 1;
        const int base = 8 + l * 8;
        const float* Wui  = (const float*)d_in[base + 0];
        const float* aSui = (const float*)d_in[base + 1];
        const float* aDui = (const float*)d_in[base + 2];
        const float* bui  = (const float*)d_in[base + 3];
        const float* Wiu  = (const float*)d_in[base + 4];
        const float* aSiu = (const float*)d_in[base + 5];
        const float* aDiu = (const float*)d_in[base + 6];
        const float* biu  = (const float*)d_in[base + 7];

        // user -> item (writes next item feats)
        run_gat(U[cur], n_user, I[cur], n_item, din,
                e_ui, e_ui + E_ui, E_ui, Wui, aSui, aDui, bui,
                HS, HD, ES, ED, MX, DN, I[nxt], stream);
        // item -> user (writes next user feats)
        run_gat(I[cur], n_item, U[cur], n_user, din,
                e_iu, e_iu + E_iu, E_iu, Wiu, aSiu, aDiu, biu,
                HS, HD, ES, ED, MX, DN, U[nxt], stream);
    }
    // after 3 layers the current features live in index 1
    float* hu = U[1];
    float* hi = I[1];

    // post MLP: out = relu(h @ W0 + b0) @ W2 + b3   (HS reused as temp)
    float* out_u = (float*)d_out;
    float* out_i = out_u + (size_t)n_user * HID;
    launch_gemm(hu, post_u[0], post_u[1], HS, n_user, GAT_N, HID, 1, stream);
    launch_gemm(HS, post_u[2], post_u[3], out_u, n_user, HID, HID, 0, stream);
    launch_gemm(hi, post_i[0], post_i[1], HS, n_item, GAT_N, HID, 1, stream);
    launch_gemm(HS, post_i[2], post_i[3], out_i, n_item, HID, HID, 0, stream);
}